// Multi_slices_forward_44040594653872
// MI455X (gfx1250) — compile-verified
//
#include <hip/hip_runtime.h>
#include <math.h>

// ---------------------------------------------------------------------------
// Multi-slice Born propagation for MI455X (gfx1250).
// 2-D FFTs implemented as tensor-core (V_WMMA_F32_16X16X4_F32) matrix FFTs:
//   512-pt DFT = F16(16x16) * X(16x32)  -> twiddle w512^(k1*n2) -> Z * G(32x32)
// Row passes stage their 16x512-complex tiles via the Tensor Data Mover
// (TENSOR_LOAD_TO_LDS / TENSOR_STORE_FROM_LDS, TENSORcnt-tracked).
// All pointwise physics fused into FFT-pass epilogues.
// ---------------------------------------------------------------------------

typedef __attribute__((ext_vector_type(2))) float v2f;
typedef __attribute__((ext_vector_type(8))) float v8f;

#define Bz   4
#define Hz   512
#define Wz   512
#define Sz   40
#define HWz  (Hz * Wz)
#define BHWz (Bz * Hz * Wz)
#define DZf  0.1f
#define PADz 32
#define OHz  (Hz - 2 * PADz)
#define OWz  (Wz - 2 * PADz)

#if defined(__gfx1250__) &&                                   \
    __has_builtin(__builtin_amdgcn_tensor_load_to_lds) &&     \
    __has_builtin(__builtin_amdgcn_tensor_store_from_lds) &&  \
    __has_builtin(__builtin_amdgcn_s_wait_tensorcnt)
#define HAVE_TDM 1
#endif

#ifdef HAVE_TDM
typedef unsigned int v4u __attribute__((ext_vector_type(4)));
typedef int v4i __attribute__((ext_vector_type(4)));
typedef int v8i __attribute__((ext_vector_type(8)));

// D# for a 2-D tile: 16 rows x 512 complex64 (8-byte elements), row stride
// 512 elements, stored linearly in LDS (l*512 + n) -- exactly tile[16][512].
__device__ __forceinline__ void tdm_desc_row_tile(unsigned long long ga,
                                                  unsigned int lds_byte,
                                                  v4u* g0, v8i* g1) {
  v4u a;
  a.x = 1u;                              // count=1 user descriptor
  a.y = lds_byte;                        // lds_addr (bytes)
  a.z = (unsigned int)ga;                // global_addr[31:0]
  a.w = ((unsigned int)(ga >> 32) & 0x01FFFFFFu) | (2u << 30);  // [56:32]|type=2
  v8i b;
  b[0] = (int)(3u << 16);                // wg_mask=0, data_size=3 (8 bytes)
  b[1] = (int)(512u << 16);              // tensor_dim0[15:0] = 512
  b[2] = (int)(16u << 16);               // tensor_dim0 hi=0 | tensor_dim1=16
  b[3] = (int)(512u << 16);              // tensor_dim1 hi=0 | tile_dim0=512
  b[4] = 16;                             // tile_dim1=16, tile_dim2=0
  b[5] = 512;                            // tensor_dim0_stride = 512 elements
  b[6] = 0;                              // stride hi | tensor_dim1_stride lo
  b[7] = 0;
  *g0 = a;
  *g1 = b;
}

__device__ __forceinline__ void tdm_load_tile(const void* g,
                                              unsigned int lds_byte) {
  v4u g0; v8i g1;
  tdm_desc_row_tile((unsigned long long)g, lds_byte, &g0, &g1);
  v4i z4 = {0, 0, 0, 0};
#if __clang_major__ >= 23
  v8i z8 = {0, 0, 0, 0, 0, 0, 0, 0};
  __builtin_amdgcn_tensor_load_to_lds(g0, g1, z4, z4, z8, 0);
#else
  __builtin_amdgcn_tensor_load_to_lds(g0, g1, z4, z4, 0);
#endif
}

__device__ __forceinline__ void tdm_store_tile(void* g, unsigned int lds_byte) {
  v4u g0; v8i g1;
  tdm_desc_row_tile((unsigned long long)g, lds_byte, &g0, &g1);
  v4i z4 = {0, 0, 0, 0};
#if __clang_major__ >= 23
  v8i z8 = {0, 0, 0, 0, 0, 0, 0, 0};
  __builtin_amdgcn_tensor_store_from_lds(g0, g1, z4, z4, z8, 0);
#else
  __builtin_amdgcn_tensor_store_from_lds(g0, g1, z4, z4, 0);
#endif
}
#endif  // HAVE_TDM

__device__ __forceinline__ float2 cmul(float2 a, float2 b) {
  return make_float2(a.x * b.x - a.y * b.y, a.x * b.y + a.y * b.x);
}

__device__ __forceinline__ v8f wmma_f32(v2f a, v2f b, v8f c) {
  // D(16x16,f32) += A(16x4,f32) * B(4x16,f32)   [V_WMMA_F32_16X16X4_F32]
  return __builtin_amdgcn_wmma_f32_16x16x4_f32(false, a, false, b, (short)0, c,
                                               false, false);
}

// One wave32 computes one 512-point complex FFT in-place on its LDS line.
// X holds x[0..511] (float2). dir = -1 fft, +1 ifft. `scale` is folded into
// the (loop-invariant, hoisted) stage-2 twiddle constants -> zero hot cost.
// Index split: n = n2 + 32*n1 (n1 in [0,16), n2 in [0,32)),
//              k = k1 + 16*k2 (k1 in [0,16), k2 in [0,32)).
__device__ __forceinline__ void wave_fft512(float2* X, int lane, float dir,
                                            float scale) {
  const int hi = lane >> 4;     // which 16-lane half
  const int ln = lane & 15;
  const float TWO_PI = 6.28318530717958648f;
  const v8f zero8 = {0.f, 0.f, 0.f, 0.f, 0.f, 0.f, 0.f, 0.f};

  // ---------------- stage 1: Y(16x32) = F16 * X ----------------
  v8f Yr[2], Yi[2];
  Yr[0] = zero8; Yr[1] = zero8; Yi[0] = zero8; Yi[1] = zero8;
#pragma unroll
  for (int kk = 0; kk < 4; ++kk) {       // K = 16 -> 4 steps of 4
    float fc[2], fs[2];
#pragma unroll
    for (int j = 0; j < 2; ++j) {        // A vgpr j: K_local = 2*hi + j
      int n1 = kk * 4 + 2 * hi + j;
      float th = dir * TWO_PI * (float)(ln * n1) * (1.f / 16.f);
      __sincosf(th, &fs[j], &fc[j]);     // F16[M=ln][n1] = cis(th)
    }
    v2f aFr = {fc[0], fc[1]};
    v2f aFi = {fs[0], fs[1]};
    v2f aFn = {-fs[0], -fs[1]};
#pragma unroll
    for (int t = 0; t < 2; ++t) {        // two 16-wide N tiles (n2)
      int n2 = t * 16 + ln;
      float2 x0 = X[n2 + 32 * (kk * 4 + 2 * hi + 0)];
      float2 x1 = X[n2 + 32 * (kk * 4 + 2 * hi + 1)];
      v2f bXr = {x0.x, x1.x};
      v2f bXi = {x0.y, x1.y};
      Yr[t] = wmma_f32(aFr, bXr, Yr[t]);   // Yr += Fr*Xr
      Yr[t] = wmma_f32(aFn, bXi, Yr[t]);   // Yr -= Fi*Xi
      Yi[t] = wmma_f32(aFr, bXi, Yi[t]);   // Yi += Fr*Xi
      Yi[t] = wmma_f32(aFi, bXr, Yi[t]);   // Yi += Fi*Xr
    }
  }

  // ---------------- twiddle w512^(k1*n2), write Z to LDS ----------------
  // C layout: vgpr v, lanes: M = v + 8*hi, N = ln (per 16-col tile).
#pragma unroll
  for (int t = 0; t < 2; ++t) {
    int n2 = t * 16 + ln;
#pragma unroll
    for (int v = 0; v < 8; ++v) {
      int k1 = v + 8 * hi;
      float s, c;
      __sincosf(dir * TWO_PI * (float)(k1 * n2) * (1.f / 512.f), &s, &c);
      float zr = Yr[t][v] * c - Yi[t][v] * s;
      float zi = Yr[t][v] * s + Yi[t][v] * c;
      X[k1 * 32 + n2] = make_float2(zr, zi);   // Z row-major [k1][n2]
    }
  }

  // ---------------- stage 2: O(16x32) = Z * (scale*G(32x32)) ----------------
  v8f Or[2], Oi[2];
  Or[0] = zero8; Or[1] = zero8; Oi[0] = zero8; Oi[1] = zero8;
#pragma unroll
  for (int kk = 0; kk < 8; ++kk) {       // K = 32 -> 8 steps of 4
    float2 za[2];
#pragma unroll
    for (int j = 0; j < 2; ++j) {        // A = Z: M=ln, K=n2
      int n2g = kk * 4 + 2 * hi + j;
      za[j] = X[ln * 32 + n2g];
    }
    v2f aZr = {za[0].x, za[1].x};
    v2f aZi = {za[0].y, za[1].y};
#pragma unroll
    for (int t = 0; t < 2; ++t) {
      int k2 = t * 16 + ln;
      float gc[2], gs[2];
#pragma unroll
      for (int j = 0; j < 2; ++j) {      // B = G: K=n2, N=k2 (scale folded in)
        int n2g = kk * 4 + 2 * hi + j;
        __sincosf(dir * TWO_PI * (float)(n2g * k2) * (1.f / 32.f), &gs[j],
                  &gc[j]);
        gc[j] *= scale;
        gs[j] *= scale;
      }
      v2f bGr = {gc[0], gc[1]};
      v2f bGi = {gs[0], gs[1]};
      v2f bGn = {-gs[0], -gs[1]};
      Or[t] = wmma_f32(aZr, bGr, Or[t]);
      Or[t] = wmma_f32(aZi, bGn, Or[t]);
      Oi[t] = wmma_f32(aZr, bGi, Oi[t]);
      Oi[t] = wmma_f32(aZi, bGr, Oi[t]);
    }
  }

  // O[k1][k2] -> X[k1 + 16*k2]
#pragma unroll
  for (int t = 0; t < 2; ++t) {
    int k2 = t * 16 + ln;
#pragma unroll
    for (int v = 0; v < 8; ++v) {
      int k1 = v + 8 * hi;
      X[k1 + 16 * k2] = make_float2(Or[t][v], Oi[t][v]);
    }
  }
}

// ---------------------------------------------------------------------------
// Generic 512-pt FFT pass over rows (rowpass=1, along W) or columns (along H).
// 16 lines per workgroup staged in LDS. Row passes use the TDM; column passes
// use coalesced VMEM with fused epilogues:
//   epi 0: out = val
//   epi 1: out = val * slab(b,h,w,slice) * DZ                 (col pass)
//   epi 2: cur = cur*prop + green * val                       (col pass)
//   epi 3: |val| -> cropped output                            (col pass)
// ---------------------------------------------------------------------------
__global__ __launch_bounds__(256) void fft512_pass(
    const float2* in, float2* out, int rowpass, float dir, float scale,
    int epi, const float2* curbuf, float2* curout, const float2* prop,
    const float2* green, const float* sre, const float* sim, const float2* sT,
    int useT, int slice, float* absout) {
  __shared__ float2 tile[16][512];   // 64 KB of the WGP's 320 KB LDS

  const int t = threadIdx.x;
  const int l0 = blockIdx.x * 16;    // first line of this workgroup
  const int b = l0 >> 9;             // 512 lines (rows or cols) per batch image
  const int cb = l0 & 511;
  const long base = (long)b * HWz;

  // -------- stage tile into LDS --------
  if (rowpass) {
#ifdef HAVE_TDM
    if (t < 32) {                    // wave 0 issues one TDM op for the tile
      tdm_load_tile(in + base + (long)cb * Wz,
                    (unsigned int)(uintptr_t)&tile[0][0]);
      __builtin_amdgcn_s_wait_tensorcnt((short)0);
    }
#else
    for (int l = 0; l < 16; ++l) {
      long rb = base + (long)(cb + l) * Wz;
      for (int n = t; n < 512; n += 256) tile[l][n] = in[rb + n];
    }
#endif
  } else {
    for (int i = 0; i < 32; ++i) {
      int idx = t + 256 * i;
      int n = idx >> 4, l = idx & 15;
      tile[l][n] = in[base + (long)n * Wz + cb + l];
    }
    if (epi == 2 && t < 16) {  // warm epilogue aux arrays (global_prefetch_b8)
      __builtin_prefetch((const char*)prop + ((long)t * Wz + cb) * 8, 0, 0);
      __builtin_prefetch((const char*)green + ((long)t * Wz + cb) * 8, 0, 0);
      __builtin_prefetch((const char*)curbuf + (base + (long)t * Wz + cb) * 8,
                         0, 0);
    }
  }
  __syncthreads();

  // -------- per-wave tensor-core FFTs (2 lines per wave32) --------
  {
    const int wave = t >> 5;
    const int lane = t & 31;
#pragma clang loop unroll(disable)
    for (int li = 0; li < 2; ++li)
      wave_fft512(&tile[wave * 2 + li][0], lane, dir, scale);
  }
  __syncthreads();

  // -------- drain tile from LDS + fused epilogue --------
  if (rowpass) {
#ifdef HAVE_TDM
    if (t < 32) {
      tdm_store_tile(out + base + (long)cb * Wz,
                     (unsigned int)(uintptr_t)&tile[0][0]);
      __builtin_amdgcn_s_wait_tensorcnt((short)0);
    }
#else
    for (int l = 0; l < 16; ++l) {
      long rb = base + (long)(cb + l) * Wz;
      for (int n = t; n < 512; n += 256) out[rb + n] = tile[l][n];
    }
#endif
  } else {
    for (int i = 0; i < 32; ++i) {
      int idx = t + 256 * i;
      int n = idx >> 4, l = idx & 15;
      int h = n, w = cb + l;
      long pg = (long)h * Wz + w;
      long g = base + pg;
      float2 v = tile[l][n];
      if (epi == 1) {                     // spatial field * slab * DZ
        float2 sl;
        if (useT) {
          sl = sT[(long)slice * BHWz + g];
        } else {
          long si = g * (long)Sz + slice;
          sl = make_float2(sre[si], sim[si]);
        }
        float2 m = cmul(v, sl);
        m.x *= DZf; m.y *= DZf;
        out[g] = m;
      } else if (epi == 2) {              // inc = inc*prop + green*scat
        float2 nc = cmul(curbuf[g], prop[pg]);
        float2 sc = cmul(green[pg], v);
        curout[g] = make_float2(nc.x + sc.x, nc.y + sc.y);
      } else if (epi == 3) {              // |.| + crop
        if (h >= PADz && h < Hz - PADz && w >= PADz && w < Wz - PADz)
          absout[(long)b * OHz * OWz + (long)(h - PADz) * OWz + (w - PADz)] =
              sqrtf(v.x * v.x + v.y * v.y);
      } else {
        out[g] = v;
      }
    }
  }
}

// prop = cis(kz*dz); green = i*prop/(2kz)*mask; pk_all = otf_abs*cis(phase-0.5kz)
// (conj(prop^25)*prop^20 = cis(-0.5*kz) folded with the OTF update)
__global__ __launch_bounds__(256) void init_fields(
    const float* kz, const float* otf_abs, const float* gmask,
    const float* otf_ph, float2* prop, float2* green, float2* pkall) {
  int i = blockIdx.x * blockDim.x + threadIdx.x;
  if (i >= HWz) return;
  float k = kz[i];
  float s, c;
  __sincosf(k * DZf, &s, &c);
  prop[i] = make_float2(c, s);
  float inv = gmask[i] / (2.0f * k);            // kz > 0 -> finite
  green[i] = make_float2(-s * inv, c * inv);    // i * cis(k*dz) / (2k) * mask
  float s2, c2;
  __sincosf(otf_ph[i] - 0.5f * k, &s2, &c2);
  pkall[i] = make_float2(otf_abs[i] * c2, otf_abs[i] * s2);
}

__global__ __launch_bounds__(256) void pack_pw(const float* re, const float* im,
                                               float2* out) {
  int i = blockIdx.x * blockDim.x + threadIdx.x;
  if (i < BHWz) out[i] = make_float2(re[i], im[i]);
}

__global__ __launch_bounds__(256) void apply_pk(float2* cur,
                                                const float2* pk) {
  int i = blockIdx.x * blockDim.x + threadIdx.x;
  if (i < BHWz) cur[i] = cmul(cur[i], pk[i & (HWz - 1)]);
}

// [B,H,W,S] (S innermost, stride-40 gather!) -> [S][B*H*W] interleaved complex.
// Reads stream contiguously per thread; writes coalesce per slice plane.
__global__ __launch_bounds__(256) void transpose_sample(const float* sre,
                                                        const float* sim,
                                                        float2* sT) {
  long i = (long)blockIdx.x * blockDim.x + threadIdx.x;
  if (i >= BHWz) return;
  long bo = i * Sz;
#pragma clang loop unroll(disable)
  for (int s = 0; s < Sz; ++s)
    sT[(long)s * BHWz + i] = make_float2(sre[bo + s], sim[bo + s]);
}

extern "C" void kernel_launch(void* const* d_in, const int* in_sizes, int n_in,
                              void* d_out, int out_size, void* d_ws,
                              size_t ws_size, hipStream_t stream) {
  const float* sre = (const float*)d_in[0];
  const float* sim = (const float*)d_in[1];
  const float* pwre = (const float*)d_in[2];
  const float* pwim = (const float*)d_in[3];
  const float* kz = (const float*)d_in[4];
  const float* otf_abs = (const float*)d_in[5];
  const float* gmask = (const float*)d_in[6];
  const float* otf_ph = (const float*)d_in[7];
  float* absout = (float*)d_out;

  char* w = (char*)d_ws;
  float2* cur = (float2*)w;   w += (size_t)BHWz * sizeof(float2);
  float2* t1 = (float2*)w;    w += (size_t)BHWz * sizeof(float2);
  float2* t2 = (float2*)w;    w += (size_t)BHWz * sizeof(float2);
  float2* prop = (float2*)w;  w += (size_t)HWz * sizeof(float2);
  float2* green = (float2*)w; w += (size_t)HWz * sizeof(float2);
  float2* pkall = (float2*)w; w += (size_t)HWz * sizeof(float2);
  size_t base_need = (size_t)(3 * BHWz + 3 * HWz) * sizeof(float2);
  float2* sT = nullptr;
  int useT = 0;
  if (ws_size >= base_need + (size_t)Sz * BHWz * sizeof(float2)) {
    sT = (float2*)w;   // avoid the stride-40 gather: ~53 GB -> 1.3 GB one-time
    useT = 1;
  }

  const float INV512 = 1.0f / 512.0f;
  const dim3 fgrid(128), fblk(256);   // (B*512 lines)/16 per workgroup

  init_fields<<<(HWz + 255) / 256, 256, 0, stream>>>(kz, otf_abs, gmask,
                                                     otf_ph, prop, green,
                                                     pkall);
  if (useT)
    transpose_sample<<<(BHWz + 255) / 256, 256, 0, stream>>>(sre, sim, sT);

  // inc0 = fft2(planewave)
  pack_pw<<<(BHWz + 255) / 256, 256, 0, stream>>>(pwre, pwim, t2);
  fft512_pass<<<fgrid, fblk, 0, stream>>>(t2, t1, 1, -1.f, 1.f, 0, nullptr,
                                          nullptr, nullptr, nullptr, nullptr,
                                          nullptr, nullptr, 0, 0, nullptr);
  fft512_pass<<<fgrid, fblk, 0, stream>>>(t1, cur, 0, -1.f, 1.f, 0, nullptr,
                                          nullptr, nullptr, nullptr, nullptr,
                                          nullptr, nullptr, 0, 0, nullptr);

  // 40-slice Born recurrence
  for (int s = 0; s < Sz; ++s) {
    fft512_pass<<<fgrid, fblk, 0, stream>>>(   // row ifft (TDM staged)
        cur, t1, 1, +1.f, INV512, 0, nullptr, nullptr, nullptr, nullptr,
        nullptr, nullptr, nullptr, 0, 0, nullptr);
    fft512_pass<<<fgrid, fblk, 0, stream>>>(   // col ifft, * slab*dz
        t1, t2, 0, +1.f, INV512, 1, nullptr, nullptr, nullptr, nullptr, sre,
        sim, sT, useT, s, nullptr);
    fft512_pass<<<fgrid, fblk, 0, stream>>>(   // row fft (TDM staged)
        t2, t1, 1, -1.f, 1.f, 0, nullptr, nullptr, nullptr, nullptr, nullptr,
        nullptr, nullptr, 0, 0, nullptr);
    fft512_pass<<<fgrid, fblk, 0, stream>>>(   // col fft, cur=cur*prop+green*v
        t1, cur, 0, -1.f, 1.f, 2, cur, cur, prop, green, nullptr, nullptr,
        nullptr, 0, 0, nullptr);
  }

  // focus/pupil: cur *= conj(prop^25)*prop^20*otf ; out = |ifft2(cur)| cropped
  apply_pk<<<(BHWz + 255) / 256, 256, 0, stream>>>(cur, pkall);
  fft512_pass<<<fgrid, fblk, 0, stream>>>(cur, t1, 1, +1.f, INV512, 0, nullptr,
                                          nullptr, nullptr, nullptr, nullptr,
                                          nullptr, nullptr, 0, 0, nullptr);
  fft512_pass<<<fgrid, fblk, 0, stream>>>(t1, t2, 0, +1.f, INV512, 3, nullptr,
                                          nullptr, nullptr, nullptr, nullptr,
                                          nullptr, nullptr, 0, 0, absout);
}